// LorentzRGCNLayer_26680336843175
// MI455X (gfx1250) — compile-verified
//
#include <hip/hip_runtime.h>
#include <math.h>

#define CURV   0.01f
#define SQRT_C 0.1f
#define EPSF   1e-7f
#define DDIM   128

typedef __attribute__((ext_vector_type(2))) float v2f;
typedef __attribute__((ext_vector_type(8))) float v8f;

__device__ __forceinline__ float wave_sum32(float v) {
#pragma unroll
  for (int off = 16; off > 0; off >>= 1)
    v += __shfl_xor(v, off, 32);
  return v;
}

__device__ __forceinline__ float artanh_c(float x) {
  x = fminf(fmaxf(x, -1.0f + EPSF), 1.0f - EPSF);
  return 0.5f * (log1pf(x) - log1pf(-x));
}

__device__ __forceinline__ float clip10(float v) {
  return fminf(fmaxf(v, -10.0f), 10.0f);
}

// ---------------- zero-fill scratch (mu + in_deg are contiguous) -------------
__global__ void k_zero(float* __restrict__ p, long long n) {
  long long i = (long long)blockIdx.x * blockDim.x + threadIdx.x;
  long long stride = (long long)gridDim.x * blockDim.x;
  for (; i < n; i += stride) p[i] = 0.0f;
}

// ---------------- h_tangent = log0(h_hyper); one wave32 per node -------------
__global__ void k_log0(const float* __restrict__ h, float* __restrict__ ht, int N) {
  int wave = (int)((blockIdx.x * blockDim.x + threadIdx.x) >> 5);
  int lane = threadIdx.x & 31;
  if (wave >= N) return;
  const float* row = h + (long long)wave * DDIM;
  float x0 = row[lane], x1 = row[lane + 32], x2 = row[lane + 64], x3 = row[lane + 96];
  float n2 = wave_sum32(x0 * x0 + x1 * x1 + x2 * x2 + x3 * x3);
  float nn = fmaxf(sqrtf(n2), EPSF);
  float s  = artanh_c(SQRT_C * nn) / (SQRT_C * nn);
  float* o = ht + (long long)wave * DDIM;
  o[lane] = s * x0; o[lane + 32] = s * x1; o[lane + 64] = s * x2; o[lane + 96] = s * x3;
}

// ---------------- float in-degree per destination ----------------------------
__global__ void k_indeg(const int* __restrict__ dst, float* __restrict__ deg, int E) {
  int i = blockIdx.x * blockDim.x + threadIdx.x;
  if (i < E) atomicAdd(&deg[dst[i]], 1.0f);
}

// ---- per-edge: msg -> exp0 -> Lorentz -> weighted centroid scatter-atomics --
__global__ void k_edges(const float* __restrict__ ht, const float* __restrict__ weight,
                        const float* __restrict__ rel_emb, const float* __restrict__ normv,
                        const float* __restrict__ deg,
                        const int* __restrict__ src, const int* __restrict__ dst,
                        const int* __restrict__ etype,
                        float* __restrict__ mu, int E) {
  int wave = (int)((blockIdx.x * blockDim.x + threadIdx.x) >> 5);
  int lane = threadIdx.x & 31;
  if (wave >= E) return;
  int s = src[wave], t = dst[wave], r = etype[wave];
  const float* x  = ht      + (long long)s * DDIM;
  const float* W  = weight  + (long long)r * DDIM;   // SI==SO==1 -> diagonal blocks
  const float* re = rel_emb + (long long)r * DDIM;
  float m0 = fmaf(x[lane],      W[lane],      re[lane]);
  float m1 = fmaf(x[lane + 32], W[lane + 32], re[lane + 32]);
  float m2 = fmaf(x[lane + 64], W[lane + 64], re[lane + 64]);
  float m3 = fmaf(x[lane + 96], W[lane + 96], re[lane + 96]);
  float n2 = wave_sum32(m0 * m0 + m1 * m1 + m2 * m2 + m3 * m3);
  float nn = fmaxf(sqrtf(n2), EPSF);
  float th = tanhf(SQRT_C * nn);
  float pscale = th / (SQRT_C * nn);              // exp0 scaling of msg
  float r2 = (th * th) / CURV;                    // ||exp0(msg)||^2 analytically
  float denom = fmaxf(1.0f - CURV * r2, EPSF);    // = max(1 - th^2, eps)
  float x0  = (1.0f + CURV * r2) / (SQRT_C * denom);
  float sps = 2.0f * pscale / denom;              // spatial scale applied to msg
  float wt  = normv[t];
  float wn  = wt / (deg[t] * wt + 1e-6f);         // wsum[n] = deg*norm + 1e-6
  float* murow = mu + (long long)t * (DDIM + 1);
  if (lane == 0) atomicAdd(&murow[0], wn * x0);
  atomicAdd(&murow[1 + lane],      wn * sps * m0);
  atomicAdd(&murow[1 + lane + 32], wn * sps * m1);
  atomicAdd(&murow[1 + lane + 64], wn * sps * m2);
  atomicAdd(&murow[1 + lane + 96], wn * sps * m3);
}

// -------- self-loop GEMM via V_WMMA_F32_16X16X4_F32 (fp32 A/B, fp32 acc) -----
// One wave per 16x16 tile: blockIdx.x = M-tile, wave-in-block = N-tile (D=128=8*16).
// Computes H@loop_weight and H@evolve_loop_weight, selects per row by in-degree.
__global__ void k_loop_wmma(const float* __restrict__ H, const float* __restrict__ Wl,
                            const float* __restrict__ We, const float* __restrict__ deg,
                            float* __restrict__ loop_msg, int N) {
  int lane = threadIdx.x & 31;
  int nt   = threadIdx.x >> 5;        // 0..7  (N-tile)
  int mt   = blockIdx.x;              // M-tile
  int half = lane >> 4;               // 0 or 1
  int l16  = lane & 15;
  int m  = mt * 16 + l16;             // A-row for this lane
  int mc = m < N ? m : (N - 1);       // clamp (rows >= N never stored)
  int n  = nt * 16 + l16;             // B/D column for this lane

  v8f c1 = {}; v8f c2 = {};
#pragma unroll 4
  for (int k = 0; k < DDIM; k += 4) {
    int ka = k + half * 2;            // 32-bit A 16x4: VGPR0={K0,K2}, VGPR1={K1,K3}
    v2f a, b1, b2;
    a.x  = H[(long long)mc * DDIM + ka];
    a.y  = H[(long long)mc * DDIM + ka + 1];
    b1.x = Wl[(long long)ka * DDIM + n];
    b1.y = Wl[(long long)(ka + 1) * DDIM + n];
    b2.x = We[(long long)ka * DDIM + n];
    b2.y = We[(long long)(ka + 1) * DDIM + n];
    c1 = __builtin_amdgcn_wmma_f32_16x16x4_f32(false, a, false, b1, (short)0, c1, false, false);
    c2 = __builtin_amdgcn_wmma_f32_16x16x4_f32(false, a, false, b2, (short)0, c2, false, false);
  }
  // D layout: lanes 0-15 -> M = r, lanes 16-31 -> M = 8 + r; N = l16
#pragma unroll
  for (int r = 0; r < 8; ++r) {
    int mm = mt * 16 + half * 8 + r;
    if (mm < N) {
      float v = (deg[mm] > 0.0f) ? c1[r] : c2[r];
      loop_msg[(long long)mm * DDIM + nt * 16 + l16] = v;
    }
  }
}

// ---- finalize: centroid -> poincare -> log0 -> clip -> +loop -> exp0 --------
__global__ void k_final(const float* __restrict__ mu, const float* __restrict__ loop_msg,
                        float* __restrict__ out, int N) {
  int wave = (int)((blockIdx.x * blockDim.x + threadIdx.x) >> 5);
  int lane = threadIdx.x & 31;
  if (wave >= N) return;
  const float* murow = mu + (long long)wave * (DDIM + 1);
  float mu0 = murow[0];
  float s0 = murow[1 + lane],      s1 = murow[1 + lane + 32];
  float s2 = murow[1 + lane + 64], s3 = murow[1 + lane + 96];
  float sq = wave_sum32(s0 * s0 + s1 * s1 + s2 * s2 + s3 * s3);
  float mink = -mu0 * mu0 + sq;
  float dm = SQRT_C * sqrtf(fmaxf(fabsf(mink), EPSF));
  float c0 = mu0 / dm;
  float pd = fmaxf(1.0f + SQRT_C * c0, EPSF);     // to_poincare denominator
  float inv = 1.0f / (dm * pd);
  float p0 = s0 * inv, p1 = s1 * inv, p2 = s2 * inv, p3 = s3 * inv;
  float pn2 = wave_sum32(p0 * p0 + p1 * p1 + p2 * p2 + p3 * p3);
  float pn = fmaxf(sqrtf(pn2), EPSF);
  float ls = artanh_c(SQRT_C * pn) / (SQRT_C * pn);
  const float* lm = loop_msg + (long long)wave * DDIM;
  float h0 = clip10(clip10(ls * p0) + lm[lane]);
  float h1 = clip10(clip10(ls * p1) + lm[lane + 32]);
  float h2 = clip10(clip10(ls * p2) + lm[lane + 64]);
  float h3 = clip10(clip10(ls * p3) + lm[lane + 96]);
  float hn2 = wave_sum32(h0 * h0 + h1 * h1 + h2 * h2 + h3 * h3);
  float hn = fmaxf(sqrtf(hn2), EPSF);
  float es = tanhf(SQRT_C * hn) / (SQRT_C * hn);  // exp0
  float* o = out + (long long)wave * DDIM;
  o[lane] = es * h0; o[lane + 32] = es * h1; o[lane + 64] = es * h2; o[lane + 96] = es * h3;
}

extern "C" void kernel_launch(void* const* d_in, const int* in_sizes, int n_in,
                              void* d_out, int out_size, void* d_ws, size_t ws_size,
                              hipStream_t stream) {
  (void)n_in; (void)out_size; (void)ws_size;
  const float* h_hyper = (const float*)d_in[0];
  const float* weight  = (const float*)d_in[1];
  const float* loop_w  = (const float*)d_in[2];
  const float* evo_w   = (const float*)d_in[3];
  const float* rel_emb = (const float*)d_in[4];
  const float* normv   = (const float*)d_in[5];
  const int*   src     = (const int*)d_in[6];
  const int*   dst     = (const int*)d_in[7];
  const int*   etype   = (const int*)d_in[8];

  int N = in_sizes[0] / DDIM;
  int E = in_sizes[6];

  float* ws       = (float*)d_ws;
  float* h_tan    = ws;                               // N * 128
  float* mu       = h_tan + (size_t)N * DDIM;         // N * 129
  float* deg      = mu + (size_t)N * (DDIM + 1);      // N   (contiguous with mu)
  float* loop_msg = deg + N;                          // N * 128

  long long zcount = (long long)N * (DDIM + 1) + N;   // mu + deg in one pass
  k_zero<<<1024, 256, 0, stream>>>(mu, zcount);

  int nodeBlocks = (N + 7) / 8;                       // 8 waves (nodes) per block
  k_log0<<<nodeBlocks, 256, 0, stream>>>(h_hyper, h_tan, N);

  k_indeg<<<(E + 255) / 256, 256, 0, stream>>>(dst, deg, E);

  k_edges<<<(E + 7) / 8, 256, 0, stream>>>(h_tan, weight, rel_emb, normv, deg,
                                           src, dst, etype, mu, E);

  int mtiles = (N + 15) / 16;                         // 3125 for N=50000
  k_loop_wmma<<<mtiles, 256, 0, stream>>>(h_tan, loop_w, evo_w, deg, loop_msg, N);

  k_final<<<nodeBlocks, 256, 0, stream>>>(mu, loop_msg, (float*)d_out, N);
}